// AgentNet_17978733101195
// MI455X (gfx1250) — compile-verified
//
#include <hip/hip_runtime.h>

// ---------------------------------------------------------------------------
// AgentNet forward on gfx1250 (MI455X), FP32 WMMA (V_WMMA_F32_16X16X4_F32).
// - gather commutes with shared-weight GEMMs: ky[idx]@wk.T == (ky@wk.T)[idx]
// - 32 agents (two 16-row M-tiles) per wave share every B fragment -> 2 WMMAs
//   per weight-fragment load (16 FLOP/B from L2 instead of 8).
// ---------------------------------------------------------------------------

typedef __attribute__((ext_vector_type(2))) float v2f;
typedef __attribute__((ext_vector_type(8))) float v8f;

#define NAGENT 50000
#define LDSTR  132            // LDS tile row stride (pad vs 64-bank LDS)

__device__ inline v8f wmma4(v2f a, v2f b, v8f c) {
  // D(16x16,f32) = A(16x4,f32) x B(4x16,f32) + C
  return __builtin_amdgcn_wmma_f32_16x16x4_f32(
      /*neg_a=*/false, a, /*neg_b=*/false, b,
      /*c_mod=*/(short)0, c, /*reuse_a=*/false, /*reuse_b=*/false);
}

// Fragment conventions (ISA 7.12.2, wave32):
//  A (16x4):  lane<16 -> row M=lane,   {K=k+0, K=k+1}
//             lane>=16-> row M=lane-16,{K=k+2, K=k+3}
//  B (4x16):  VGPR j, lane<16 -> row K=k+j,   col N=lane
//                     lane>=16-> row K=k+j+2, col N=lane-16
//  C/D(16x16):VGPR r, lane<16 -> row M=r,   col N=lane
//                     lane>=16-> row M=r+8, col N=lane-16

// Dual-M-tile GEMM, A rows from global (row-major, lda). Row indices of the
// second tile are clamped to maxrow (keeps EXEC all-ones for WMMA; clamped
// rows only affect tile-1 outputs, which the caller skips storing).
template<int NT>
__device__ inline void gemm2_a_global(const float* __restrict__ A, int lda,
                                      int maxrow,
                                      const float* __restrict__ W,
                                      v8f (&acc0)[NT], v8f (&acc1)[NT],
                                      int K, int lane) {
  const int half = lane >> 4, m = lane & 15;
  const int ld = NT * 16;
  const int r0 = m;
  const int r1 = min(m + 16, maxrow);
  #pragma unroll
  for (int t = 0; t < NT; ++t) {
    v8f z = {0.f,0.f,0.f,0.f,0.f,0.f,0.f,0.f};
    acc0[t] = z; acc1[t] = z;
  }
  #pragma unroll 2
  for (int k = 0; k < K; k += 4) {
    v2f a0, a1;
    a0.x = A[r0 * lda + k + 2*half];
    a0.y = A[r0 * lda + k + 2*half + 1];
    a1.x = A[r1 * lda + k + 2*half];
    a1.y = A[r1 * lda + k + 2*half + 1];
    #pragma unroll
    for (int t = 0; t < NT; ++t) {
      const float* Wp = W + (k + 2*half) * ld + t*16 + m;
      v2f b; b.x = Wp[0]; b.y = Wp[ld];
      acc0[t] = wmma4(a0, b, acc0[t]);
      acc1[t] = wmma4(a1, b, acc1[t]);
    }
  }
}

// Dual-M-tile GEMM, A rows from a 32 x LDSTR LDS tile.
template<int NT>
__device__ inline void gemm2_a_lds(const float* __restrict__ As,
                                   const float* __restrict__ W,
                                   v8f (&acc0)[NT], v8f (&acc1)[NT],
                                   int K, int lane) {
  const int half = lane >> 4, m = lane & 15;
  const int ld = NT * 16;
  #pragma unroll
  for (int t = 0; t < NT; ++t) {
    v8f z = {0.f,0.f,0.f,0.f,0.f,0.f,0.f,0.f};
    acc0[t] = z; acc1[t] = z;
  }
  #pragma unroll 2
  for (int k = 0; k < K; k += 4) {
    v2f a0, a1;
    a0.x = As[m * LDSTR + k + 2*half];
    a0.y = As[m * LDSTR + k + 2*half + 1];
    a1.x = As[(m + 16) * LDSTR + k + 2*half];
    a1.y = As[(m + 16) * LDSTR + k + 2*half + 1];
    #pragma unroll
    for (int t = 0; t < NT; ++t) {
      const float* Wp = W + (k + 2*half) * ld + t*16 + m;
      v2f b; b.x = Wp[0]; b.y = Wp[ld];
      acc0[t] = wmma4(a0, b, acc0[t]);
      acc1[t] = wmma4(a1, b, acc1[t]);
    }
  }
}

// Store both 8-tile D sets (32x128) + bias into a 32 x LDSTR LDS tile.
__device__ inline void store_d2_lds(const v8f (&a0)[8], const v8f (&a1)[8],
                                    const float* __restrict__ bias,
                                    float* __restrict__ Ls, int lane) {
  const int half = lane >> 4, m = lane & 15;
  #pragma unroll
  for (int t = 0; t < 8; ++t) {
    const int n = t*16 + m;
    const float b = bias[n];
    #pragma unroll
    for (int r = 0; r < 8; ++r) {
      const int mr = r + 8*half;
      Ls[mr * LDSTR + n]        = a0[t][r] + b;
      Ls[(mr + 16) * LDSTR + n] = a1[t][r] + b;
    }
  }
}

// Store both D sets + bias to global [*,128]; tile 1 only if t1ok.
__device__ inline void store_d2_global(const v8f (&a0)[8], const v8f (&a1)[8],
                                       const float* __restrict__ bias,
                                       float* __restrict__ G, int agent0,
                                       bool t1ok, int lane) {
  const int half = lane >> 4, m = lane & 15;
  #pragma unroll
  for (int t = 0; t < 8; ++t) {
    const int n = t*16 + m;
    const float b = bias[n];
    #pragma unroll
    for (int r = 0; r < 8; ++r) {
      const int mr = r + 8*half;
      G[(size_t)(agent0 + mr) * 128 + n] = a0[t][r] + b;
      if (t1ok) G[(size_t)(agent0 + 16 + mr) * 128 + n] = a1[t][r] + b;
    }
  }
}

// ---------------------------------------------------------------------------
// Kernel 0: transpose in_proj_w (384x128) -> wqT,wkT,wvT each [k][c]
__global__ __launch_bounds__(256) void k_transpose_inproj(
    const float* __restrict__ in_proj_w, float* __restrict__ wT) {
  int i = blockIdx.x * blockDim.x + threadIdx.x;
  if (i >= 3 * 128 * 128) return;
  const int c = i & 127;
  const int k = (i >> 7) & 127;
  const int mmat = i >> 14;            // 0=q, 1=k, 2=v
  wT[i] = in_proj_w[((mmat << 7) + c) * 128 + k];
}

// ---------------------------------------------------------------------------
// Kernel A: one wave32 per 32-agent slab:
//   hidden=relu(obs@enc_w+b); msg/ky; q=msg@wq.T; kk=ky@wk.T; vv=msg@wv.T
__global__ __launch_bounds__(32) void k_encode_project(
    const float* __restrict__ obs,
    const float* __restrict__ enc_w, const float* __restrict__ enc_b,
    const float* __restrict__ msg_w, const float* __restrict__ msg_b,
    const float* __restrict__ key_w, const float* __restrict__ key_b,
    const float* __restrict__ wqT, const float* __restrict__ wkT,
    const float* __restrict__ wvT, const float* __restrict__ in_proj_b,
    float* __restrict__ hiddenG, float* __restrict__ qG,
    float* __restrict__ kkG, float* __restrict__ vvG) {
  __shared__ float hidS[32 * LDSTR];
  __shared__ float msgS[32 * LDSTR];
  __shared__ float kyS [32 * LDSTR];
  const int lane = threadIdx.x;
  const int half = lane >> 4, m = lane & 15;
  const int agent0 = blockIdx.x * 32;
  const int maxrow = NAGENT - 1 - agent0;          // >= 15 always
  const bool t1ok = (agent0 + 16) < NAGENT;        // wave-uniform
  v8f a0[8], a1[8];

  // hidden = relu(obs @ enc_w + enc_b), K=64
  gemm2_a_global<8>(obs + (size_t)agent0 * 64, 64, maxrow, enc_w,
                    a0, a1, 64, lane);
  #pragma unroll
  for (int t = 0; t < 8; ++t) {
    const int n = t*16 + m;
    const float b = enc_b[n];
    #pragma unroll
    for (int r = 0; r < 8; ++r) {
      const int mr = r + 8*half;
      const float h0 = fmaxf(a0[t][r] + b, 0.f);
      const float h1 = fmaxf(a1[t][r] + b, 0.f);
      hidS[mr * LDSTR + n]        = h0;
      hidS[(mr + 16) * LDSTR + n] = h1;
      hiddenG[(size_t)(agent0 + mr) * 128 + n] = h0;
      if (t1ok) hiddenG[(size_t)(agent0 + 16 + mr) * 128 + n] = h1;
    }
  }
  // single-wave workgroup: compiler-inserted s_wait_dscnt covers LDS RAW

  gemm2_a_lds<8>(hidS, msg_w, a0, a1, 128, lane);      // msg
  store_d2_lds(a0, a1, msg_b, msgS, lane);

  gemm2_a_lds<8>(hidS, key_w, a0, a1, 128, lane);      // ky
  store_d2_lds(a0, a1, key_b, kyS, lane);

  gemm2_a_lds<8>(msgS, wqT, a0, a1, 128, lane);        // q = msg@wq.T + bq
  store_d2_global(a0, a1, in_proj_b + 0, qG, agent0, t1ok, lane);

  gemm2_a_lds<8>(kyS, wkT, a0, a1, 128, lane);         // kk = ky@wk.T + bk
  store_d2_global(a0, a1, in_proj_b + 128, kkG, agent0, t1ok, lane);

  gemm2_a_lds<8>(msgS, wvT, a0, a1, 128, lane);        // vv = msg@wv.T + bv
  store_d2_global(a0, a1, in_proj_b + 256, vvG, agent0, t1ok, lane);
}

// ---------------------------------------------------------------------------
// Kernel B: attention. One wave32 per agent; kk/vv gathers are L2-resident
// (kk+vv = 51 MB << 192 MB L2).
__global__ __launch_bounds__(256) void k_attention(
    const float* __restrict__ q, const float* __restrict__ kk,
    const float* __restrict__ vv, const int* __restrict__ nidx,
    float* __restrict__ ctx) {
  const int wave = (int)((blockIdx.x * blockDim.x + threadIdx.x) >> 5);
  const int lane = threadIdx.x & 31;
  if (wave >= NAGENT) return;

  const float4 qc = *(const float4*)&q[(size_t)wave * 128 + lane * 4];
  const int nbr = nidx[wave * 16 + (lane & 15)];

  float myscore = -3.4e38f;
  #pragma unroll
  for (int k = 0; k < 16; ++k) {
    const int nb = __shfl(nbr, k, 32);
    const float4 kc = *(const float4*)&kk[(size_t)nb * 128 + lane * 4];
    float p = qc.x*kc.x + qc.y*kc.y + qc.z*kc.z + qc.w*kc.w;
    #pragma unroll
    for (int off = 16; off >= 1; off >>= 1) p += __shfl_xor(p, off, 32);
    if (lane == k) myscore = p * 0.08838834764831845f;   // 1/sqrt(128)
  }

  // softmax over lanes 0..15
  float s = (lane < 16) ? myscore : -3.4e38f;
  float mx = s;
  #pragma unroll
  for (int off = 16; off >= 1; off >>= 1) mx = fmaxf(mx, __shfl_xor(mx, off, 32));
  float e = (lane < 16) ? __expf(s - mx) : 0.f;
  float sum = e;
  #pragma unroll
  for (int off = 16; off >= 1; off >>= 1) sum += __shfl_xor(sum, off, 32);
  const float w = e / sum;

  float4 c = {0.f, 0.f, 0.f, 0.f};
  #pragma unroll
  for (int k = 0; k < 16; ++k) {
    const float wk = __shfl(w, k, 32);
    const int nb = __shfl(nbr, k, 32);
    const float4 vc = *(const float4*)&vv[(size_t)nb * 128 + lane * 4];
    c.x += wk * vc.x; c.y += wk * vc.y; c.z += wk * vc.z; c.w += wk * vc.w;
  }
  *(float4*)&ctx[(size_t)wave * 128 + lane * 4] = c;
}

// ---------------------------------------------------------------------------
// Kernel C: attn_out = ctx@out_w+out_b; new_hidden = hidden+attn_out;
//           logits = new_hidden@dec_w+dec_b; out = softmax(logits)
__global__ __launch_bounds__(32) void k_output_decode(
    const float* __restrict__ ctx, const float* __restrict__ hiddenG,
    const float* __restrict__ out_w, const float* __restrict__ out_b,
    const float* __restrict__ dec_w, const float* __restrict__ dec_b,
    float* __restrict__ out) {
  __shared__ float nhS[32 * LDSTR];
  __shared__ float lgS[32 * 34];
  const int lane = threadIdx.x;
  const int half = lane >> 4, m = lane & 15;
  const int agent0 = blockIdx.x * 32;
  const int maxrow = NAGENT - 1 - agent0;
  v8f a0[8], a1[8];

  gemm2_a_global<8>(ctx + (size_t)agent0 * 128, 128, maxrow, out_w,
                    a0, a1, 128, lane);
  #pragma unroll
  for (int t = 0; t < 8; ++t) {
    const int n = t*16 + m;
    const float b = out_b[n];
    #pragma unroll
    for (int r = 0; r < 8; ++r) {
      const int mr = r + 8*half;
      const int h1row = min(agent0 + 16 + mr, NAGENT - 1);
      nhS[mr * LDSTR + n] =
          a0[t][r] + b + hiddenG[(size_t)(agent0 + mr) * 128 + n];
      nhS[(mr + 16) * LDSTR + n] =
          a1[t][r] + b + hiddenG[(size_t)h1row * 128 + n];
    }
  }

  v8f l0[2], l1[2];
  gemm2_a_lds<2>(nhS, dec_w, l0, l1, 128, lane);       // logits: 32x32
  #pragma unroll
  for (int t = 0; t < 2; ++t) {
    const int n = t*16 + m;
    const float b = dec_b[n];
    #pragma unroll
    for (int r = 0; r < 8; ++r) {
      const int mr = r + 8*half;
      lgS[mr * 34 + n]        = l0[t][r] + b;
      lgS[(mr + 16) * 34 + n] = l1[t][r] + b;
    }
  }

  // row softmax: each lane owns one of the 32 rows
  const int row = lane;
  if (agent0 + row < NAGENT) {
    const float* rp = &lgS[row * 34];
    float mx = -3.4e38f;
    #pragma unroll
    for (int c = 0; c < 32; ++c) mx = fmaxf(mx, rp[c]);
    float sum = 0.f;
    #pragma unroll
    for (int c = 0; c < 32; ++c) sum += __expf(rp[c] - mx);
    const float inv = 1.f / sum;
    float* o = &out[(size_t)(agent0 + row) * 32];
    #pragma unroll
    for (int c = 0; c < 32; ++c) o[c] = __expf(rp[c] - mx) * inv;
  }
}

// ---------------------------------------------------------------------------
extern "C" void kernel_launch(void* const* d_in, const int* in_sizes, int n_in,
                              void* d_out, int out_size, void* d_ws, size_t ws_size,
                              hipStream_t stream) {
  const float* obs       = (const float*)d_in[0];
  const int*   nidx      = (const int*)  d_in[1];
  const float* enc_w     = (const float*)d_in[2];
  const float* enc_b     = (const float*)d_in[3];
  const float* msg_w     = (const float*)d_in[4];
  const float* msg_b     = (const float*)d_in[5];
  const float* key_w     = (const float*)d_in[6];
  const float* key_b     = (const float*)d_in[7];
  const float* in_proj_w = (const float*)d_in[8];
  const float* in_proj_b = (const float*)d_in[9];
  const float* out_w     = (const float*)d_in[10];
  const float* out_b     = (const float*)d_in[11];
  const float* dec_w     = (const float*)d_in[12];
  const float* dec_b     = (const float*)d_in[13];
  float* o = (float*)d_out;

  float* ws = (float*)d_ws;
  const size_t NH = (size_t)NAGENT * 128;
  float* hidden = ws + 0 * NH;
  float* qb     = ws + 1 * NH;
  float* kkb    = ws + 2 * NH;
  float* vvb    = ws + 3 * NH;
  float* ctxb   = ws + 4 * NH;
  float* wT     = ws + 5 * NH;       // 3*128*128 floats: wqT|wkT|wvT
  float* wqT = wT, *wkT = wT + 128*128, *wvT = wT + 2*128*128;

  const int nblk = (NAGENT + 31) / 32;   // 1563

  k_transpose_inproj<<<(3*128*128 + 255) / 256, 256, 0, stream>>>(in_proj_w, wT);

  k_encode_project<<<nblk, 32, 0, stream>>>(
      obs, enc_w, enc_b, msg_w, msg_b, key_w, key_b,
      wqT, wkT, wvT, in_proj_b, hidden, qb, kkb, vvb);

  k_attention<<<(NAGENT * 32 + 255) / 256, 256, 0, stream>>>(
      qb, kkb, vvb, nidx, ctxb);

  k_output_decode<<<nblk, 32, 0, stream>>>(
      ctxb, hidden, out_w, out_b, dec_w, dec_b, o);
}